// Encoder_5282809774688
// MI455X (gfx1250) — compile-verified
//
#include <hip/hip_runtime.h>
#include <hip/hip_bf16.h>

typedef __attribute__((ext_vector_type(2))) float v2f;
typedef __attribute__((ext_vector_type(8))) float v8f;

#define B_    2
#define C_    32
#define T_    64
#define H_    64
#define W_    64
#define HW_   (H_ * W_)          // 4096
#define N_    512
#define CF_   32
#define CH_   16
#define OUTT_ 400
#define L_    (OUTT_ * N_)       // 204800
#define TILES_PER_B_ (L_ / 16)   // 12800
#define NTILES_ (B_ * TILES_PER_B_) // 25600
#define BN_COUNT_ ((float)(B_ * L_))  // 409600

__device__ __forceinline__ void lerpT(int t_out, int& i0, int& i1, float& w) {
    float pos = (float)t_out * (float(T_ - 1) / float(OUTT_ - 1));
    int a = (int)floorf(pos);
    if (a > T_ - 1) a = T_ - 1;
    if (a < 0) a = 0;
    i0 = a;
    i1 = (a + 1 < T_) ? a + 1 : T_ - 1;
    w  = pos - (float)a;
}

// ---------------- kernel 0: zero the stats accumulators ----------------
__global__ void k_zero(float* __restrict__ p, int n) {
    int i = blockIdx.x * blockDim.x + threadIdx.x;
    if (i < n) p[i] = 0.0f;
}

// ---------------- kernel 1: result head, pre-lerp ----------------
// r_small[b,t,hw] = sum_c Wr[c]*x[b,c,t,hw] + br
__global__ void k_result_small(const float* __restrict__ x,
                               const float* __restrict__ Wr,
                               const float* __restrict__ br,
                               float* __restrict__ r_small) {
    int idx = blockIdx.x * blockDim.x + threadIdx.x;
    if (idx >= B_ * T_ * HW_) return;
    int hw = idx % HW_;
    int t  = (idx / HW_) % T_;
    int b  = idx / (T_ * HW_);
    const float* xp = x + ((size_t)(b * C_) * T_ + t) * HW_ + hw;
    float acc = br[0];
    #pragma unroll
    for (int c = 0; c < C_; ++c)
        acc = fmaf(Wr[c], xp[(size_t)c * T_ * HW_], acc);
    r_small[idx] = acc;
}

// ---------------- kernel 2: result head, T-lerp 64 -> 400 ----------------
__global__ void k_result_lerp(const float* __restrict__ r_small,
                              float* __restrict__ out_result) {
    int idx = blockIdx.x * blockDim.x + threadIdx.x;
    if (idx >= B_ * OUTT_ * HW_) return;
    int hw    = idx % HW_;
    int t_out = (idx / HW_) % OUTT_;
    int b     = idx / (OUTT_ * HW_);
    int i0, i1; float w;
    lerpT(t_out, i0, i1, w);
    float v0 = r_small[(b * T_ + i0) * HW_ + hw];
    float v1 = r_small[(b * T_ + i1) * HW_ + hw];
    out_result[idx] = v0 + (v1 - v0) * w;
}

// ---------------- kernel 3: gathered feature head (pre-lerp) ----------------
// f_small[(b*T+t)*N+n][cf] = sum_c Wf[cf,c]*x[b,c,t,coord(b,n)] + bf[cf]
__global__ void k_feat_gather(const float* __restrict__ x,
                              const int* __restrict__ coord,
                              const float* __restrict__ Wf,
                              const float* __restrict__ bf,
                              float* __restrict__ f_small) {
    __shared__ float sWf[CF_ * C_];
    __shared__ float sbf[CF_];
    for (int i = threadIdx.x; i < CF_ * C_; i += blockDim.x) sWf[i] = Wf[i];
    if (threadIdx.x < CF_) sbf[threadIdx.x] = bf[threadIdx.x];
    __syncthreads();
    int idx = blockIdx.x * blockDim.x + threadIdx.x;   // over B*T*N
    if (idx >= B_ * T_ * N_) return;
    int n = idx % N_;
    int t = (idx / N_) % T_;
    int b = idx / (T_ * N_);
    int hh = coord[(b * N_ + n) * 2 + 0];
    int ww = coord[(b * N_ + n) * 2 + 1];
    int hw = hh * W_ + ww;
    float xv[C_];
    const float* xp = x + ((size_t)(b * C_) * T_ + t) * HW_ + hw;
    #pragma unroll
    for (int c = 0; c < C_; ++c) xv[c] = xp[(size_t)c * T_ * HW_];
    float* op = f_small + (size_t)idx * CF_;
    #pragma unroll
    for (int cf = 0; cf < CF_; ++cf) {
        float acc = sbf[cf];
        #pragma unroll
        for (int c = 0; c < C_; ++c) acc = fmaf(sWf[cf * C_ + c], xv[c], acc);
        op[cf] = acc;
    }
}

// ---------------- WMMA helpers (V_WMMA_F32_16X16X4_F32) ----------------
// A (16x4 f32): lanes 0-15: M=lane, K={0,1}; lanes 16-31: M=lane-16, K={2,3}
// B (4x16 f32): lanes 0-15: N=lane, K={0,1}; lanes 16-31: N=lane-16, K={2,3}
// C/D: VGPR j: lanes 0-15 -> M=j, N=lane; lanes 16-31 -> M=j+8, N=lane-16
__device__ __forceinline__ void load_a_frags(const float* __restrict__ Wmat,
                                             int ncol, v2f* a, int nk, int lane) {
    int m    = lane & 15;
    int koff = (lane >> 4) << 1;   // 0 or 2
    #pragma unroll
    for (int k = 0; k < nk; ++k)
        a[k] = *(const v2f*)(Wmat + m * ncol + k * 4 + koff);
}

// h-tile = W1 (16x32) x g-block (32x16), g lerped on the fly from f_small
__device__ __forceinline__ v8f compute_h_tile(const float* __restrict__ f_small,
                                              const v2f a1[8],
                                              int b, int t0, int t1, float w,
                                              int n0, int lane) {
    int col  = lane & 15;
    int koff = (lane >> 4) << 1;
    int n = n0 + col;
    const float* p0 = f_small + ((size_t)(b * T_ + t0) * N_ + n) * CF_;
    const float* p1 = f_small + ((size_t)(b * T_ + t1) * N_ + n) * CF_;
    v8f acc = {};
    #pragma unroll
    for (int k = 0; k < 8; ++k) {
        int c0 = k * 4 + koff;
        v2f g0 = *(const v2f*)(p0 + c0);
        v2f g1 = *(const v2f*)(p1 + c0);
        v2f bfrag = g0 + (g1 - g0) * w;
        acc = __builtin_amdgcn_wmma_f32_16x16x4_f32(
            false, a1[k], false, bfrag, (short)0, acc, false, false);
    }
    return acc;
}

// ---------------- kernel 4: BN statistics pass (WMMA, no h spill) ----------------
__global__ void k_proj_stats(const float* __restrict__ f_small,
                             const float* __restrict__ W1,
                             const float* __restrict__ b1,
                             float* __restrict__ gsum,
                             float* __restrict__ gsumsq) {
    __shared__ float lsum[CH_], lsq[CH_];
    if (threadIdx.x < CH_) { lsum[threadIdx.x] = 0.0f; lsq[threadIdx.x] = 0.0f; }
    __syncthreads();

    int lane   = threadIdx.x & 31;
    int wave   = (blockIdx.x * blockDim.x + threadIdx.x) >> 5;
    int nwaves = (gridDim.x * blockDim.x) >> 5;
    int hi     = (lane >> 4) * 8;

    v2f a1[8];
    load_a_frags(W1, CF_, a1, 8, lane);
    float b1r[8];
    #pragma unroll
    for (int j = 0; j < 8; ++j) b1r[j] = b1[j + hi];

    float s[8], sq[8];
    #pragma unroll
    for (int j = 0; j < 8; ++j) { s[j] = 0.0f; sq[j] = 0.0f; }

    for (int tile = wave; tile < NTILES_; tile += nwaves) {
        int b     = tile / TILES_PER_B_;
        int lt    = tile % TILES_PER_B_;
        int t_out = lt >> 5;            // 32 tiles of 16 per t (N=512)
        int n0    = (lt & 31) << 4;
        int i0, i1; float w;
        lerpT(t_out, i0, i1, w);
        v8f h = compute_h_tile(f_small, a1, b, i0, i1, w, n0, lane);
        #pragma unroll
        for (int j = 0; j < 8; ++j) {
            float v = h[j] + b1r[j];
            s[j]  += v;
            sq[j] += v * v;
        }
    }
    #pragma unroll
    for (int j = 0; j < 8; ++j) {
        atomicAdd(&lsum[j + hi], s[j]);
        atomicAdd(&lsq[j + hi], sq[j]);
    }
    __syncthreads();
    if (threadIdx.x < CH_) {
        atomicAdd(&gsum[threadIdx.x],   lsum[threadIdx.x]);
        atomicAdd(&gsumsq[threadIdx.x], lsq[threadIdx.x]);
    }
}

// ---------------- kernel 5: finalize BN scale/shift ----------------
__global__ void k_finalize(const float* __restrict__ gsum,
                           const float* __restrict__ gsumsq,
                           const float* __restrict__ gamma,
                           const float* __restrict__ beta,
                           float* __restrict__ scale,
                           float* __restrict__ shift) {
    int i = threadIdx.x;
    if (i < CH_) {
        float inv = 1.0f / BN_COUNT_;
        float mu  = gsum[i] * inv;
        float var = gsumsq[i] * inv - mu * mu;
        var = var > 0.0f ? var : 0.0f;
        float sc = gamma[i] * rsqrtf(var + 1e-5f);
        scale[i] = sc;
        shift[i] = beta[i] - mu * sc;
    }
}

// ---------------- kernel 6: projection output pass (2x WMMA chains) ----------------
__global__ void k_proj_out(const float* __restrict__ f_small,
                           const float* __restrict__ W1,
                           const float* __restrict__ b1,
                           const float* __restrict__ W2,
                           const float* __restrict__ b2,
                           const float* __restrict__ scale,
                           const float* __restrict__ shift,
                           float* __restrict__ out) {
    __shared__ float tile_lds[8][16 * 17];   // one 16x16 tile per wave, pad 17
    int lane = threadIdx.x & 31;
    int warp = threadIdx.x >> 5;
    int tile = blockIdx.x * (blockDim.x >> 5) + warp;  // grid sized exactly
    int col  = lane & 15;
    int hi   = (lane >> 4) * 8;
    int koff = (lane >> 4) << 1;

    v2f a1[8], a2[4];
    load_a_frags(W1, CF_, a1, 8, lane);
    load_a_frags(W2, CH_, a2, 4, lane);
    float b1r[8], scr[8], shr[8], b2r[8];
    #pragma unroll
    for (int j = 0; j < 8; ++j) {
        b1r[j] = b1[j + hi];
        scr[j] = scale[j + hi];
        shr[j] = shift[j + hi];
        b2r[j] = b2[j + hi];
    }

    int b     = tile / TILES_PER_B_;
    int lt    = tile % TILES_PER_B_;
    int t_out = lt >> 5;
    int n0    = (lt & 31) << 4;
    int i0, i1; float w;
    lerpT(t_out, i0, i1, w);

    v8f h = compute_h_tile(f_small, a1, b, i0, i1, w, n0, lane);

    // BN + ReLU, stage through LDS to re-fragment rows(channels) as K for WMMA #2
    float* my = &tile_lds[warp][0];
    #pragma unroll
    for (int j = 0; j < 8; ++j) {
        float v = h[j] + b1r[j];
        v = fmaf(v, scr[j], shr[j]);
        v = v > 0.0f ? v : 0.0f;
        my[(j + hi) * 17 + col] = v;
    }
    __syncthreads();

    v8f acc = {};
    #pragma unroll
    for (int k = 0; k < 4; ++k) {
        int c0 = k * 4 + koff;
        v2f bfrag;
        bfrag.x = my[c0 * 17 + col];
        bfrag.y = my[(c0 + 1) * 17 + col];
        acc = __builtin_amdgcn_wmma_f32_16x16x4_f32(
            false, a2[k], false, bfrag, (short)0, acc, false, false);
    }

    int l = t_out * N_ + n0 + col;
    #pragma unroll
    for (int j = 0; j < 8; ++j) {
        int ch = j + hi;
        out[((size_t)b * CH_ + ch) * L_ + l] = acc[j] + b2r[j];
    }
}

// ---------------- host launcher ----------------
extern "C" void kernel_launch(void* const* d_in, const int* in_sizes, int n_in,
                              void* d_out, int out_size, void* d_ws, size_t ws_size,
                              hipStream_t stream) {
    (void)in_sizes; (void)n_in; (void)out_size; (void)ws_size;
    const float* x     = (const float*)d_in[0];
    const int*   coord = (const int*)  d_in[1];
    const float* Wf    = (const float*)d_in[2];
    const float* bf    = (const float*)d_in[3];
    const float* Wr    = (const float*)d_in[4];
    const float* br    = (const float*)d_in[5];
    const float* W1    = (const float*)d_in[6];
    const float* b1    = (const float*)d_in[7];
    const float* gamma = (const float*)d_in[8];
    const float* beta  = (const float*)d_in[9];
    const float* W2    = (const float*)d_in[10];
    const float* b2    = (const float*)d_in[11];

    float* out        = (float*)d_out;
    float* out_proj   = out;                                   // [2,16,204800]
    float* out_result = out + (size_t)B_ * CH_ * L_;           // [2,1,400,64,64]

    float* ws       = (float*)d_ws;
    float* r_small  = ws;                                      // 524288 floats
    float* f_small  = ws + (size_t)B_ * T_ * HW_;              // 2097152 floats
    float* gsum     = f_small + (size_t)B_ * T_ * N_ * CF_;    // 16
    float* gsumsq   = gsum + CH_;                              // 16
    float* scale    = gsum + 2 * CH_;                          // 16
    float* shift    = gsum + 3 * CH_;                          // 16

    k_zero<<<1, 64, 0, stream>>>(gsum, 4 * CH_);

    k_result_small<<<(B_ * T_ * HW_) / 256, 256, 0, stream>>>(x, Wr, br, r_small);
    k_result_lerp<<<(B_ * OUTT_ * HW_) / 256, 256, 0, stream>>>(r_small, out_result);

    k_feat_gather<<<(B_ * T_ * N_) / 256, 256, 0, stream>>>(x, coord, Wf, bf, f_small);

    k_proj_stats<<<256, 256, 0, stream>>>(f_small, W1, b1, gsum, gsumsq);
    k_finalize<<<1, 32, 0, stream>>>(gsum, gsumsq, gamma, beta, scale, shift);
    k_proj_out<<<NTILES_ / 8, 256, 0, stream>>>(f_small, W1, b1, W2, b2,
                                                scale, shift, out_proj);
}